// GRUDecoder_6725918786243
// MI455X (gfx1250) — compile-verified
//
#include <hip/hip_runtime.h>
#include <math.h>

// Problem constants (match reference)
#define B_   128
#define L_   32
#define V_   10000
#define IC_  2048
#define WD_  512
#define H_   512
#define G3H  1536          // 3*H
#define DLW  2560          // W_ih row length (IC+WD)
#define T_   31            // L-1

typedef __attribute__((ext_vector_type(2))) float v2f;
typedef __attribute__((ext_vector_type(4))) float v4f;
typedef __attribute__((ext_vector_type(8))) float v8f;

__device__ __forceinline__ v8f wmma_f32(v2f a, v2f b, v8f c) {
  // V_WMMA_F32_16X16X4_F32 : exact fp32 D = A(16x4) x B(4x16) + C
  return __builtin_amdgcn_wmma_f32_16x16x4_f32(false, a, false, b, (short)0, c,
                                               false, false);
}

__device__ __forceinline__ float sigmoidf_(float x) {
  return 1.0f / (1.0f + expf(-x));
}

// ---------------------------------------------------------------------------
// Kernel 1: stable descending sort of cap_lens (rank-based, 128 elems),
// gather captions, and emit the three integer tuple outputs (as float).
// ---------------------------------------------------------------------------
__global__ void k_sort(const int* __restrict__ cap_lens,
                       const int* __restrict__ captions,
                       int* __restrict__ order_ws, int* __restrict__ len_ws,
                       int* __restrict__ caps_ws,
                       float* __restrict__ out_caps, float* __restrict__ out_lens,
                       float* __restrict__ out_order) {
  __shared__ int lens[B_];
  const int i = threadIdx.x;
  lens[i] = cap_lens[i];
  __syncthreads();
  const int li = lens[i];
  int rank = 0;
  for (int j = 0; j < B_; ++j) {
    const int lj = lens[j];
    rank += (lj > li) || (lj == li && j < i);   // stable argsort(-lens)
  }
  order_ws[rank] = i;
  len_ws[rank]   = li - 1;
  out_lens[rank]  = (float)(li - 1);
  out_order[rank] = (float)i;
  for (int l = 0; l < L_; ++l) {
    const int c = captions[i * L_ + l];
    caps_ws[rank * L_ + l]  = c;
    out_caps[rank * L_ + l] = (float)c;
  }
}

// ---------------------------------------------------------------------------
// Kernel 2: generic fp32 WMMA GEMM, C[m,n] = sum_k A[row(m), k]*W[n, koff+k]
//   A rows optionally gathered through rowidx (used for image_code[order]).
//   Requires M % 128 == 0 or M == 128, K/16 even; N may be ragged.
//   mode 0: C[m*ldc+n] = acc + bias[n]
//   mode 1: FC epilogue: m=(t,b), mask t<lens[b], store preds[b,t,n] (else 0)
// Block: 128 threads / 4 waves; 128x64 tile; wave w owns rows [32w,32w+32)
// and all 64 cols (2x4 wmma tiles -> 0.75 LDS frag loads per wmma).
// Software-pipelined ping-pong staging with compile-time buffer indices.
// ---------------------------------------------------------------------------
__global__ void k_gemm(const float* __restrict__ A, const int* __restrict__ rowidx,
                       int lda,
                       const float* __restrict__ W, int ldw, int koff, int K,
                       const float* __restrict__ bias,
                       float* __restrict__ C, int ldc, int M, int N,
                       int mode, const int* __restrict__ lens) {
  __shared__ float As[2][128][17];  // [buf][m][k]
  __shared__ float Bs[2][64][17];   // [buf][n][k]
  const int tid  = threadIdx.x;
  const int m0   = blockIdx.y * 128, n0 = blockIdx.x * 64;
  const int w    = tid >> 5, lane = tid & 31;
  const int half = lane >> 4, lr = lane & 15;
  const int wr   = w * 32;

  // hoisted per-thread staging pointers: A 4 float4 (128x16), B 2 float4 (64x16)
  const float* apt[4];
  int am[4], ak4[4];
#pragma unroll
  for (int i = 0; i < 4; ++i) {
    const int e = i * 128 + tid;          // float4 slot
    am[i]  = e >> 2;
    ak4[i] = (e & 3) * 4;
    const int gm = m0 + am[i];            // always < M by launch contract
    const int r  = rowidx ? rowidx[gm] : gm;
    apt[i] = A + (size_t)r * lda + ak4[i];
  }
  const float* bpt[2];
  bool bvld[2];
  int  bm[2], bk4[2];
#pragma unroll
  for (int i = 0; i < 2; ++i) {
    const int e = i * 128 + tid;
    bm[i]  = e >> 2;
    bk4[i] = (e & 3) * 4;
    const int gn = n0 + bm[i];
    bvld[i] = gn < N;
    const int wrow = bvld[i] ? gn : (N - 1);   // clamp: load valid, zero later
    bpt[i] = W + (size_t)wrow * ldw + koff + bk4[i];
  }

  const v8f vz8 = {0.f, 0.f, 0.f, 0.f, 0.f, 0.f, 0.f, 0.f};
  const v4f vz4 = {0.f, 0.f, 0.f, 0.f};
  v8f acc[2][4];
#pragma unroll
  for (int a = 0; a < 2; ++a)
#pragma unroll
    for (int b = 0; b < 4; ++b) acc[a][b] = vz8;

  auto stage = [&](int k0, int buf) {
#pragma unroll
    for (int i = 0; i < 4; ++i) {
      const v4f va = *(const v4f*)(apt[i] + k0);
#pragma unroll
      for (int q = 0; q < 4; ++q) As[buf][am[i]][ak4[i] + q] = va[q];
    }
#pragma unroll
    for (int i = 0; i < 2; ++i) {
      v4f vb = *(const v4f*)(bpt[i] + k0);
      if (!bvld[i]) vb = vz4;
      __builtin_prefetch(bpt[i] + k0 + 16, 0, 1);   // global_prefetch_b8
#pragma unroll
      for (int q = 0; q < 4; ++q) Bs[buf][bm[i]][bk4[i] + q] = vb[q];
    }
  };

  auto compute = [&](int buf) {
#pragma unroll
    for (int kk = 0; kk < 4; ++kk) {
      const int kb = kk * 4 + half * 2;   // f32 A/B frag: v0=K{0,2}, v1=K{1,3}
      v2f a0, a1;
      a0[0] = As[buf][wr + lr][kb];       a0[1] = As[buf][wr + lr][kb + 1];
      a1[0] = As[buf][wr + 16 + lr][kb];  a1[1] = As[buf][wr + 16 + lr][kb + 1];
#pragma unroll
      for (int nt = 0; nt < 4; ++nt) {
        v2f b;
        b[0] = Bs[buf][nt * 16 + lr][kb];
        b[1] = Bs[buf][nt * 16 + lr][kb + 1];
        acc[0][nt] = wmma_f32(a0, b, acc[0][nt]);
        acc[1][nt] = wmma_f32(a1, b, acc[1][nt]);
      }
    }
  };

  // software pipeline, NC even; buffer indices are compile-time constants
  const int NC = K / 16;
  stage(0, 0);
  __syncthreads();
  for (int c = 0; c < NC - 2; c += 2) {
    stage((c + 1) * 16, 1);
    compute(0);
    __syncthreads();
    stage((c + 2) * 16, 0);
    compute(1);
    __syncthreads();
  }
  stage((NC - 1) * 16, 1);
  compute(0);
  __syncthreads();
  compute(1);

  // epilogue: C VGPR v holds rows {v, v+8} split by half-wave
#pragma unroll
  for (int mt = 0; mt < 2; ++mt)
#pragma unroll
    for (int nt = 0; nt < 4; ++nt)
#pragma unroll
      for (int v = 0; v < 8; ++v) {
        const int m = m0 + wr + mt * 16 + v + half * 8;
        const int n = n0 + nt * 16 + lr;
        if (n >= N) continue;
        const float val = acc[mt][nt][v] + (bias ? bias[n] : 0.f);
        if (mode == 0) {
          C[(size_t)m * ldc + n] = val;
        } else {
          const int t = m >> 7, b = m & 127;      // m = t*128 + b
          const float o = (t < lens[b]) ? val : 0.f;
          C[((size_t)b * T_ + t) * V_ + n] = o;   // preds[b, t, n]
        }
      }
}

// ---------------------------------------------------------------------------
// Recurrence GEMM pass: acc += Arows(128 x K) @ Wrows(32 x K)^T
// A rows optionally gathered via caption tokens (embedding fused into GEMM;
// token lookup hoisted out of the K loop). Pipelined LDS double buffering
// with compile-time buffer indices (K/16 even).
// Block: 128 threads / 4 waves; wave w owns rows [32w,32w+32) x 32 cols.
// ---------------------------------------------------------------------------
__device__ __forceinline__ void rec_gemm_pass(
    v8f (&acc)[2][2], float (&As)[2][128][17], float (&Bs)[2][32][17],
    const float* __restrict__ A, const int* __restrict__ caps, int t, int lda,
    const float* __restrict__ W, int ldw, int koff, int wrow0, int K) {
  const int tid  = threadIdx.x;
  const int w    = tid >> 5, lane = tid & 31;
  const int half = lane >> 4, lr = lane & 15;
  const int wr   = w * 32;

  // hoisted per-thread staging pointers: 4 float4 for A (128x16), 1 for B (32x16)
  const float* apt[4];
  int am[4], ak4[4];
#pragma unroll
  for (int i = 0; i < 4; ++i) {
    const int e = i * 128 + tid;
    am[i]  = e >> 2;
    ak4[i] = (e & 3) * 4;
    const int r = caps ? caps[am[i] * L_ + t] : am[i];   // embedding gather
    apt[i] = A + (size_t)r * lda + ak4[i];
  }
  const int bn  = tid >> 2;
  const int bk4 = (tid & 3) * 4;
  const float* bpt = W + (size_t)(wrow0 + bn) * ldw + koff + bk4;

  auto stage = [&](int k0, int buf) {
#pragma unroll
    for (int i = 0; i < 4; ++i) {
      const v4f va = *(const v4f*)(apt[i] + k0);
#pragma unroll
      for (int q = 0; q < 4; ++q) As[buf][am[i]][ak4[i] + q] = va[q];
    }
    const v4f vb = *(const v4f*)(bpt + k0);
    __builtin_prefetch(bpt + k0 + 16, 0, 1);
#pragma unroll
    for (int q = 0; q < 4; ++q) Bs[buf][bn][bk4 + q] = vb[q];
  };

  auto compute = [&](int buf) {
#pragma unroll
    for (int kk = 0; kk < 4; ++kk) {
      const int kb = kk * 4 + half * 2;
      v2f a0, a1, b0, b1;
      a0[0] = As[buf][wr + lr][kb];       a0[1] = As[buf][wr + lr][kb + 1];
      a1[0] = As[buf][wr + 16 + lr][kb];  a1[1] = As[buf][wr + 16 + lr][kb + 1];
      b0[0] = Bs[buf][lr][kb];            b0[1] = Bs[buf][lr][kb + 1];
      b1[0] = Bs[buf][16 + lr][kb];       b1[1] = Bs[buf][16 + lr][kb + 1];
      acc[0][0] = wmma_f32(a0, b0, acc[0][0]);
      acc[0][1] = wmma_f32(a0, b1, acc[0][1]);
      acc[1][0] = wmma_f32(a1, b0, acc[1][0]);
      acc[1][1] = wmma_f32(a1, b1, acc[1][1]);
    }
  };

  const int NC = K / 16;   // even for K in {512}
  stage(0, 0);
  __syncthreads();
  for (int c = 0; c < NC - 2; c += 2) {
    stage((c + 1) * 16, 1);
    compute(0);
    __syncthreads();
    stage((c + 2) * 16, 0);
    compute(1);
    __syncthreads();
  }
  stage((NC - 1) * 16, 1);
  compute(0);
  __syncthreads();
  compute(1);
  // no trailing barrier needed: next pass stages buf 0 (reads of buf 0 were
  // fenced by the barrier above) and fences buf 1 before touching it.
}

// ---------------------------------------------------------------------------
// Kernel 3: one GRU step. Block owns 32 hidden columns; all gate
// pre-activations stay register-resident (same lane map across passes).
//   gi (input gates)  = gi_img (precomputed, incl. b_ih) + emb @ W_ih_word^T
//   gh (hidden gates) = h @ W_hh^T + b_hh
//   r,z = sigmoid(gi+gh);  n = tanh(i_n + r*h_n);  h' = (1-z)*n + z*h
// ---------------------------------------------------------------------------
__global__ void k_gru_step(const float* __restrict__ h_prev,
                           float* __restrict__ h_out,
                           const int* __restrict__ caps_ws,
                           const float* __restrict__ gi_img,
                           const float* __restrict__ embed_w,
                           const float* __restrict__ W_ih,
                           const float* __restrict__ W_hh,
                           const float* __restrict__ b_hh, int t) {
  __shared__ float As[2][128][17];
  __shared__ float Bs[2][32][17];
  const int tid  = threadIdx.x;
  const int w    = tid >> 5, lane = tid & 31;
  const int half = lane >> 4, lr = lane & 15;
  const int j0   = blockIdx.x * 32;

  const v8f vz = {0.f, 0.f, 0.f, 0.f, 0.f, 0.f, 0.f, 0.f};
  v8f accZ[2][2], accR[2][2], accNg[2][2], accNi[2][2];
#pragma unroll
  for (int a = 0; a < 2; ++a)
#pragma unroll
    for (int b = 0; b < 2; ++b) {
      accZ[a][b] = vz; accR[a][b] = vz; accNg[a][b] = vz; accNi[a][b] = vz;
    }

  // z gates (W rows 512+j): hidden part + word part into one accumulator
  rec_gemm_pass(accZ, As, Bs, h_prev, nullptr, t, H_, W_hh, H_, 0, 512 + j0, H_);
  rec_gemm_pass(accZ, As, Bs, embed_w, caps_ws, t, WD_, W_ih, DLW, IC_, 512 + j0, WD_);
  // r gates (W rows 0+j)
  rec_gemm_pass(accR, As, Bs, h_prev, nullptr, t, H_, W_hh, H_, 0, j0, H_);
  rec_gemm_pass(accR, As, Bs, embed_w, caps_ws, t, WD_, W_ih, DLW, IC_, j0, WD_);
  // n gates (W rows 1024+j): hidden and input parts kept separate
  rec_gemm_pass(accNg, As, Bs, h_prev, nullptr, t, H_, W_hh, H_, 0, 1024 + j0, H_);
  rec_gemm_pass(accNi, As, Bs, embed_w, caps_ws, t, WD_, W_ih, DLW, IC_, 1024 + j0, WD_);

  // fused gates + h update (register resident)
  const int wr = w * 32;
#pragma unroll
  for (int nt = 0; nt < 2; ++nt) {
    const int j  = j0 + nt * 16 + lr;
    const float br = b_hh[j];
    const float bz = b_hh[512 + j];
    const float bn = b_hh[1024 + j];
#pragma unroll
    for (int mt = 0; mt < 2; ++mt)
#pragma unroll
      for (int v = 0; v < 8; ++v) {
        const int m = wr + mt * 16 + v + half * 8;     // batch row
        const float* gi = gi_img + (size_t)m * G3H;    // includes b_ih
        const float z   = sigmoidf_(gi[512 + j]  + accZ[mt][nt][v] + bz);
        const float r   = sigmoidf_(gi[j]        + accR[mt][nt][v] + br);
        const float ghn = accNg[mt][nt][v] + bn;
        const float inn = gi[1024 + j] + accNi[mt][nt][v];
        const float n   = tanhf(inn + r * ghn);
        const float hp  = h_prev[(size_t)m * H_ + j];
        h_out[(size_t)m * H_ + j] = (1.f - z) * n + z * hp;
      }
  }
}

// ---------------------------------------------------------------------------
extern "C" void kernel_launch(void* const* d_in, const int* in_sizes, int n_in,
                              void* d_out, int out_size, void* d_ws, size_t ws_size,
                              hipStream_t stream) {
  const float* image_code = (const float*)d_in[0];
  const int*   captions   = (const int*)d_in[1];
  const int*   cap_lens   = (const int*)d_in[2];
  const float* embed_w    = (const float*)d_in[3];
  const float* W_ih       = (const float*)d_in[4];
  const float* W_hh       = (const float*)d_in[5];
  const float* b_ih       = (const float*)d_in[6];
  const float* b_hh       = (const float*)d_in[7];
  const float* fc_w       = (const float*)d_in[8];
  const float* fc_b       = (const float*)d_in[9];
  const float* init_w     = (const float*)d_in[10];
  const float* init_b     = (const float*)d_in[11];

  // workspace layout (~9.2 MB)
  int*   order_ws = (int*)d_ws;                 // 128
  int*   len_ws   = order_ws + B_;              // 128
  int*   caps_ws  = len_ws + B_;                // 128*32
  float* h0       = (float*)(caps_ws + B_ * L_);// 128*512
  float* gi_img   = h0 + B_ * H_;               // 128*1536
  float* outs     = gi_img + (size_t)B_ * G3H;  // 31*128*512 (h_t for all t)

  // output layout: preds (B,T,V) f32, then captions_s, lengths, order as f32
  float* preds     = (float*)d_out;
  float* out_caps  = preds + (size_t)B_ * T_ * V_;
  float* out_lens  = out_caps + B_ * L_;
  float* out_order = out_lens + B_;

  // 1) stable sort + gathers + integer outputs
  k_sort<<<1, B_, 0, stream>>>(cap_lens, captions, order_ws, len_ws, caps_ws,
                               out_caps, out_lens, out_order);

  // 2) h0 = image_code[order] @ init_w^T + init_b    (128x512, K=2048)
  k_gemm<<<dim3(H_ / 64, 1), 128, 0, stream>>>(
      image_code, order_ws, IC_, init_w, IC_, 0, IC_, init_b,
      h0, H_, B_, H_, 0, nullptr);

  // 3) gi_img = image_code[order] @ W_ih[:, :2048]^T + b_ih   (128x1536, K=2048)
  k_gemm<<<dim3(G3H / 64, 1), 128, 0, stream>>>(
      image_code, order_ws, IC_, W_ih, DLW, 0, IC_, b_ih,
      gi_img, G3H, B_, G3H, 0, nullptr);

  // 4) GRU recurrence: 31 sequential steps, 16 blocks x 128 threads each
  for (int t = 0; t < T_; ++t) {
    const float* hp = (t == 0) ? h0 : (outs + (size_t)(t - 1) * B_ * H_);
    float* ho = outs + (size_t)t * B_ * H_;
    k_gru_step<<<H_ / 32, 128, 0, stream>>>(hp, ho, caps_ws, gi_img, embed_w,
                                            W_ih, W_hh, b_hh, t);
  }

  // 5) FC: preds = mask(outs @ fc_w^T + fc_b), stored as (B,T,V)
  k_gemm<<<dim3((V_ + 63) / 64, (T_ * B_) / 128), 128, 0, stream>>>(
      outs, nullptr, H_, fc_w, H_, 0, H_, fc_b,
      preds, V_, T_ * B_, V_, 1, len_ws);
}